// PointPillarsScatter_59081570124839
// MI455X (gfx1250) — compile-verified
//
#include <hip/hip_runtime.h>
#include <cstdint>

// ---------------------------------------------------------------------------
// PointPillarsScatter for MI455X (gfx1250).
// Pure data movement: 0 FLOPs, ~235 MB of traffic -> HBM-roofline bound
// (~10 us at 23.3 TB/s). WMMA is intentionally NOT used: there is no matrix
// product in this op. Optimization = coalesced b128 stores with NT cache
// policy + keeping the 3.4 MB inverse map and 12.3 MB feature table L2-hot.
// ---------------------------------------------------------------------------

// Native clang vectors (required by __builtin_nontemporal_store; HIP's
// float4/int4 are wrapper classes and are rejected by the builtin).
typedef float f32x4 __attribute__((ext_vector_type(4)));
typedef int   i32x4 __attribute__((ext_vector_type(4)));

namespace {

constexpr int kB      = 4;
constexpr int kC      = 64;
constexpr int kNZ     = 1;
constexpr int kNY     = 496;
constexpr int kNX     = 432;
constexpr int kNspace = kNZ * kNY * kNX;           // 214272 (divisible by 4)

// --- Path 1: inverse-map gather (preferred) --------------------------------

// map[b*nspace + idx] = voxel id, or -1 if empty.
__global__ void init_map_kernel(i32x4* __restrict__ map, int n4) {
  int t = blockIdx.x * blockDim.x + threadIdx.x;
  if (t < n4) {
    i32x4 m = {-1, -1, -1, -1};
    __builtin_nontemporal_store(m, map + t);  // streaming init, keep L2 clean
  }
}

__global__ void build_map_kernel(const int* __restrict__ coords,
                                 int* __restrict__ map, int nvox) {
  int t = blockIdx.x * blockDim.x + threadIdx.x;
  if (t < nvox) {
    // coords row (b, z, y, x) is 16B-aligned -> one b128 load.
    const i32x4 c = reinterpret_cast<const i32x4*>(coords)[t];
    const int idx = c.y * kNZ + c.z * kNX + c.w;   // matches reference indexing
    map[c.x * kNspace + idx] = t;
  }
}

// Output-stationary: thread t covers 4 consecutive spatial positions of one
// (b, c) plane. Writes are perfectly coalesced b128 NT stores; map reads are
// coalesced and L2-resident; feature gathers are random 4B but the whole
// 12.3 MB feature table lives in the 192 MB L2 after first touch.
__global__ void emit_kernel(const float* __restrict__ feat,
                            const int* __restrict__ map,
                            float* __restrict__ out, int n4) {
  int t = blockIdx.x * blockDim.x + threadIdx.x;
  if (t >= n4) return;
  const int e      = t << 2;                 // flat output element index
  const int plane  = e / kNspace;            // = b*kC + c   (const-div -> mulhi)
  const int within = e - plane * kNspace;    // spatial idx, multiple of 4
  const int c      = plane & (kC - 1);
  const int b      = plane >> 6;

  const i32x4 m =
      reinterpret_cast<const i32x4*>(map)[(b * kNspace + within) >> 2];

  f32x4 v;
  v.x = (m.x < 0) ? 0.0f : feat[m.x * kC + c];
  v.y = (m.y < 0) ? 0.0f : feat[m.y * kC + c];
  v.z = (m.z < 0) ? 0.0f : feat[m.z * kC + c];
  v.w = (m.w < 0) ? 0.0f : feat[m.w * kC + c];

  // Streaming 219 MB output: NT store policy so it does not evict the
  // L2-resident map/features (global_store_b128 ... th:TH_STORE_NT).
  __builtin_nontemporal_store(v, reinterpret_cast<f32x4*>(out + e));
}

// --- Path 2: fallback fill + scatter (only if workspace is too small) ------

__global__ void fill_zero_kernel(f32x4* __restrict__ out, int n4) {
  int t = blockIdx.x * blockDim.x + threadIdx.x;
  if (t < n4) {
    f32x4 z = {0.0f, 0.0f, 0.0f, 0.0f};
    __builtin_nontemporal_store(z, out + t);
  }
}

__global__ void scatter_kernel(const float* __restrict__ feat,
                               const int* __restrict__ coords,
                               float* __restrict__ out, int total) {
  int t = blockIdx.x * blockDim.x + threadIdx.x;   // t = vox*64 + c
  if (t < total) {
    const int vox = t >> 6;
    const int c   = t & (kC - 1);
    const i32x4 cr = reinterpret_cast<const i32x4*>(coords)[vox];
    const int idx = cr.y * kNZ + cr.z * kNX + cr.w;
    out[(cr.x * kC + c) * kNspace + idx] = feat[t];
  }
}

}  // namespace

extern "C" void kernel_launch(void* const* d_in, const int* in_sizes, int n_in,
                              void* d_out, int out_size, void* d_ws, size_t ws_size,
                              hipStream_t stream) {
  const float* feat   = reinterpret_cast<const float*>(d_in[0]);
  const int*   coords = reinterpret_cast<const int*>(d_in[1]);
  float*       out    = reinterpret_cast<float*>(d_out);

  const int nvox  = in_sizes[1] / 4;                      // 48000
  const int outN4 = out_size >> 2;                        // out_size % 4 == 0 here

  const size_t mapBytes = (size_t)kB * kNspace * sizeof(int);  // 3.43 MB

  constexpr int kBlk = 256;  // 8 wave32 waves per workgroup

  if (ws_size >= mapBytes) {
    int* map = reinterpret_cast<int*>(d_ws);

    const int mapN4 = (kB * kNspace) >> 2;                // 214272
    init_map_kernel<<<(mapN4 + kBlk - 1) / kBlk, kBlk, 0, stream>>>(
        reinterpret_cast<i32x4*>(map), mapN4);

    build_map_kernel<<<(nvox + kBlk - 1) / kBlk, kBlk, 0, stream>>>(
        coords, map, nvox);

    emit_kernel<<<(outN4 + kBlk - 1) / kBlk, kBlk, 0, stream>>>(
        feat, map, out, outN4);
  } else {
    // Fallback: zero-fill then scatter (more HBM traffic, still correct).
    fill_zero_kernel<<<(outN4 + kBlk - 1) / kBlk, kBlk, 0, stream>>>(
        reinterpret_cast<f32x4*>(out), outN4);

    const int total = nvox * kC;
    scatter_kernel<<<(total + kBlk - 1) / kBlk, kBlk, 0, stream>>>(
        feat, coords, out, total);
  }
}